// seismic_cas_farword_28046136442964
// MI455X (gfx1250) — compile-verified
//
#include <hip/hip_runtime.h>

typedef __attribute__((ext_vector_type(2))) float v2f;
typedef __attribute__((ext_vector_type(8))) float v8f;

namespace {
constexpr int NZc = 150, NXc = 300, NTc = 1000, NSc = 4, NRc = 400;
constexpr float DXf = 10.0f, DTf = 0.001f;
constexpr int NZP = NZc + 8, NXP = NXc + 8;   // zero halo of 4 on each side
constexpr int BZT = 32, BXT = 64;             // block tile (z x x)
constexpr int LR = BZT + 8, LC = BXT + 8;     // LDS tile 40 x 72
constexpr int LCV = LC / 4;                   // float4 columns per LDS row (18)
}

// C[|d|]/DX^2 banded-stencil weight, 0 outside |d|<=4
__device__ __forceinline__ float stencil_w(int d) {
  const float inv = 1.0f / (DXf * DXf);
  int a = d < 0 ? -d : d;
  float c;
  switch (a) {
    case 0: c = -205.0f / 72.0f; break;
    case 1: c =    8.0f /  5.0f; break;
    case 2: c =   -1.0f /  5.0f; break;
    case 3: c =    8.0f / 315.0f; break;
    case 4: c =   -1.0f / 560.0f; break;
    default: c = 0.0f; break;
  }
  return c * inv;
}

__global__ void seis_zero_kernel(float* p, int n) {
  int i = blockIdx.x * blockDim.x + threadIdx.x;
  if (i < n) p[i] = 0.0f;
}

// v2p (padded NZP x NXP) = (v_map * DT)^2 in interior
__global__ void seis_v2_kernel(const float* __restrict__ v_map, float* __restrict__ v2p) {
  int i = blockIdx.x * blockDim.x + threadIdx.x;
  if (i >= NZc * NXc) return;
  int z = i / NXc, x = i % NXc;
  float v = v_map[i] * DTf;
  v2p[(size_t)(z + 4) * NXP + (x + 4)] = v * v;
}

// One time step: u_next = 2*u_cur - u_prev + v2dt2 * Lap(u_cur), source folded in.
// u_next may alias u_prev (per-element read-before-write within one thread).
__global__ __launch_bounds__(256) void seis_step_wmma_kernel(
    const float* __restrict__ u_cur, const float* u_prev, float* u_next,
    const float* __restrict__ v2p, const float* __restrict__ s_amp,
    const int* __restrict__ s_loc, int t) {
  __shared__ __align__(16) float lds[LR * LC];

  const int bz = blockIdx.x;             // 0..4
  const int bx = blockIdx.y;             // 0..4
  const int s  = blockIdx.z;             // shot
  const int z0b = 4 + bz * BZT;
  const int x0b = 4 + bx * BXT;
  const float* uc = u_cur + (size_t)s * NZP * NXP;
  const float* up = u_prev + (size_t)s * NZP * NXP;
  float*       un = u_next + (size_t)s * NZP * NXP;

  // prefetch u_prev tile into caches (gfx1250 global_prefetch_b8)
  {
    int pr = threadIdx.x >> 3;           // 0..31
    int pc = (threadIdx.x & 7) * 8;      // 0..56
    int gz = z0b + pr; if (gz > NZP - 1) gz = NZP - 1;
    int gx = x0b + pc; if (gx > NXP - 1) gx = NXP - 1;
    __builtin_prefetch(&up[(size_t)gz * NXP + gx], 0, 3);
  }

  // ---- async stage of u_cur tile (40 x 72, 4-wide halo) straight into LDS ----
  // Direct L2->LDS path (ASYNCcnt), b128 granularity; both global rows
  // (1232 B stride) and the LDS tile are 16B aligned. Column clamp at
  // float4 granularity still lands inside the zero halo, so valid outputs
  // only ever see correct values.
  {
    const unsigned lds_base = (unsigned)(uintptr_t)lds;
    for (int i = threadIdx.x; i < LR * LCV; i += 256) {
      int r = i / LCV, c4 = (i - r * LCV) * 4;
      int gz = z0b - 4 + r;  if (gz > NZP - 1) gz = NZP - 1;
      int gx = x0b - 4 + c4; if (gx > NXP - 4) gx = NXP - 4;
      unsigned ldsoff = lds_base + (unsigned)((r * LC + c4) * 4);
      unsigned long long ga =
          (unsigned long long)(uintptr_t)&uc[(size_t)gz * NXP + gx];
      asm volatile("global_load_async_to_lds_b128 %0, %1, off"
                   :: "v"(ldsoff), "v"(ga) : "memory");
    }
    asm volatile("s_wait_asynccnt 0x0" ::: "memory");
  }
  __syncthreads();

  const int wave = threadIdx.x >> 5;     // 0..7
  const int lane = threadIdx.x & 31;
  const int wz = wave & 1;               // 2 z-subtiles
  const int wx = wave >> 1;              // 4 x-subtiles
  const int m  = lane & 15;              // M (A/const) or N (B/const) index
  const int hi = lane >> 4;              // K-pair selector

  // constant banded-matrix fragments: w(K - m - 4); identical per-lane values
  // serve as B for the x-pass and as A for the z-pass (band symmetry).
  v2f wf[6];
#pragma unroll
  for (int j = 0; j < 6; ++j) {
    int kb = j * 4 + hi * 2;
    wf[j].x = stencil_w(kb - m - 4);
    wf[j].y = stencil_w(kb + 1 - m - 4);
  }

  v8f acc = {};
  // x-direction: A = u rows (16x4 chunks), B = const band
#pragma unroll
  for (int j = 0; j < 6; ++j) {
    int kb = j * 4 + hi * 2;
    int r = wz * 16 + 4 + m;             // LDS row (z)
    int c = wx * 16 + kb;                // LDS col (x - 4 origin)
    v2f a = *(const v2f*)(&lds[r * LC + c]);   // 8B aligned: c even, LC even
    acc = __builtin_amdgcn_wmma_f32_16x16x4_f32(false, a, false, wf[j],
                                                (short)0, acc, false, false);
  }
  // z-direction: A = const band, B = u columns (4x16 chunks)
#pragma unroll
  for (int j = 0; j < 6; ++j) {
    int kb = j * 4 + hi * 2;
    int r = wz * 16 + kb;                // LDS row (z - 4 origin)
    int c = wx * 16 + 4 + m;             // LDS col (x)
    v2f b;
    b.x = lds[r * LC + c];
    b.y = lds[(r + 1) * LC + c];
    acc = __builtin_amdgcn_wmma_f32_16x16x4_f32(false, wf[j], false, b,
                                                (short)0, acc, false, false);
  }

  // combine: u_next = 2*u_cur - u_prev + v2dt2 * Lap ; fold source injection
  const int n  = m;                      // column within subtile
  const int gx = x0b + wx * 16 + n;
  const int ssz = s_loc[2 * s + 0] + 4;
  const int ssx = s_loc[2 * s + 1] + 4;
  const float amp = s_amp[s * NTc + t];

  const bool full = ((z0b + wz * 16 + 16) <= (4 + NZc)) &&
                    ((x0b + wx * 16 + 16) <= (4 + NXc));  // wave-uniform
  if (full) {
#pragma unroll
    for (int v = 0; v < 8; ++v) {
      int mm = v + hi * 8;               // row within subtile (C/D layout)
      int gz = z0b + wz * 16 + mm;
      size_t idx = (size_t)gz * NXP + gx;
      float v2 = v2p[idx];
      float val = 2.0f * lds[(wz * 16 + 4 + mm) * LC + (wx * 16 + 4 + n)]
                  - up[idx] + v2 * acc[v];
      if (gz == ssz && gx == ssx) val += v2 * amp;  // src_scale == v2dt2[sz,sx]
      un[idx] = val;
    }
  } else {
#pragma unroll
    for (int v = 0; v < 8; ++v) {
      int mm = v + hi * 8;
      int gz = z0b + wz * 16 + mm;
      if (gz < 4 + NZc && gx < 4 + NXc) {
        size_t idx = (size_t)gz * NXP + gx;
        float v2 = v2p[idx];
        float val = 2.0f * lds[(wz * 16 + 4 + mm) * LC + (wx * 16 + 4 + n)]
                    - up[idx] + v2 * acc[v];
        if (gz == ssz && gx == ssx) val += v2 * amp;
        un[idx] = val;
      }
    }
  }
}

__global__ void seis_record_kernel(const float* __restrict__ u,
                                   const int* __restrict__ r_loc,
                                   float* __restrict__ out, int t) {
  int i = blockIdx.x * blockDim.x + threadIdx.x;
  if (i >= NSc * NRc) return;
  int s = i / NRc, r = i % NRc;
  int rz = r_loc[2 * r + 0] + 4;
  int rx = r_loc[2 * r + 1] + 4;
  out[(size_t)s * NTc * NRc + (size_t)t * NRc + r] =
      u[(size_t)s * NZP * NXP + (size_t)rz * NXP + rx];
}

extern "C" void kernel_launch(void* const* d_in, const int* in_sizes, int n_in,
                              void* d_out, int out_size, void* d_ws, size_t ws_size,
                              hipStream_t stream) {
  const float* v_map = (const float*)d_in[0];  // (NZ, NX)
  const float* s_amp = (const float*)d_in[1];  // (NS, NT)
  const int*   s_loc = (const int*)d_in[2];    // (NS, 2)
  const int*   r_loc = (const int*)d_in[3];    // (NR, 2)
  float* out = (float*)d_out;                  // (NS, NT, NR)

  const size_t ufield = (size_t)NSc * NZP * NXP;   // 194,656 floats
  float* u_a = (float*)d_ws;
  float* u_b = u_a + ufield;
  float* v2p = u_b + ufield;                       // NZP*NXP floats

  // zero-init wavefields each call (halo stays zero -> reference's zero pad)
  {
    int ztot = (int)(2 * ufield);
    seis_zero_kernel<<<(ztot + 255) / 256, 256, 0, stream>>>(u_a, ztot);
  }
  seis_v2_kernel<<<(NZc * NXc + 255) / 256, 256, 0, stream>>>(v_map, v2p);

  float* u_prev = u_a;
  float* u_cur  = u_b;
  dim3 grid((NZc + BZT - 1) / BZT, (NXc + BXT - 1) / BXT, NSc);  // (5,5,4)
  for (int t = 0; t < NTc; ++t) {
    // writes u_next into u_prev's buffer (safe elementwise aliasing)
    seis_step_wmma_kernel<<<grid, 256, 0, stream>>>(u_cur, u_prev, u_prev,
                                                    v2p, s_amp, s_loc, t);
    float* u_next = u_prev;
    seis_record_kernel<<<(NSc * NRc + 255) / 256, 256, 0, stream>>>(u_next,
                                                                    r_loc, out, t);
    u_prev = u_cur;
    u_cur  = u_next;
  }
}